// Balanced_EDMLoss_39539468927258
// MI455X (gfx1250) — compile-verified
//
#include <hip/hip_runtime.h>
#include <hip/hip_bf16.h>
#include <math.h>

// ---------------------------------------------------------------------------
// Balanced EMD loss, B rows x 10 cols, fp32. Memory-bound streaming reduction.
// CDNA5 path: GLOBAL_LOAD_ASYNC_TO_LDS_B64 staging + s_wait_asynccnt.
// ---------------------------------------------------------------------------

#define TILE_ROWS 256
#define TILE_FLOATS (TILE_ROWS * 10)   // 2560 floats per input per tile
#define MAX_BLOCKS 8192

#if defined(__has_builtin)
#if __has_builtin(__builtin_amdgcn_global_load_async_to_lds_b64)
#define HAVE_ASYNC_LDS 1
#endif
#endif

typedef int v2i __attribute__((ext_vector_type(2)));
typedef __attribute__((address_space(1))) v2i* gv2i_p;   // global int2*
typedef __attribute__((address_space(3))) v2i* lv2i_p;   // LDS int2*

__device__ __forceinline__ void copy_b64_to_lds(void* lds_dst, const void* g_src) {
#if defined(HAVE_ASYNC_LDS)
    __builtin_amdgcn_global_load_async_to_lds_b64(
        (gv2i_p)g_src, (lv2i_p)lds_dst, /*imm offset*/0, /*cpol*/0);
#else
    *(float2*)lds_dst = *(const float2*)g_src;
#endif
}

__device__ __forceinline__ void wait_async_zero() {
#if defined(HAVE_ASYNC_LDS)
#if defined(__has_builtin) && __has_builtin(__builtin_amdgcn_s_wait_asynccnt)
    __builtin_amdgcn_s_wait_asynccnt(0);
#else
    asm volatile("s_wait_asynccnt 0" ::: "memory");
#endif
#endif
}

__global__ __launch_bounds__(TILE_ROWS) void emd_partial_kernel(
    const float* __restrict__ gpt, const float* __restrict__ gpe,
    float* __restrict__ partials, int B, int numTiles) {
    __shared__ float spt[TILE_FLOATS];
    __shared__ float spe[TILE_FLOATS];
    __shared__ float swave[8];

    const int tid = threadIdx.x;
    float lacc = 0.0f;

    for (int tile = blockIdx.x; tile < numTiles; tile += gridDim.x) {
        const size_t baseF = (size_t)tile * TILE_FLOATS;
        const int rowsHere = min(TILE_ROWS, B - tile * TILE_ROWS);

        if (rowsHere == TILE_ROWS) {
            // Full tile: 1280 float2 per input, 5 coalesced async b64 copies/lane.
            const float2* g2t = (const float2*)(gpt + baseF);
            const float2* g2e = (const float2*)(gpe + baseF);
            float2* s2t = (float2*)spt;
            float2* s2e = (float2*)spe;
#pragma unroll
            for (int i = 0; i < 5; ++i) {
                const int idx = i * TILE_ROWS + tid;
                copy_b64_to_lds(&s2t[idx], &g2t[idx]);
                copy_b64_to_lds(&s2e[idx], &g2e[idx]);
            }
            wait_async_zero();
        } else {
            // Tail tile: guarded scalar staging.
            const int nf = rowsHere * 10;
            for (int i = tid; i < nf; i += TILE_ROWS) {
                spt[i] = gpt[baseF + i];
                spe[i] = gpe[baseF + i];
            }
        }
        __syncthreads();

        if (tid < rowsHere) {
            const float* pt = &spt[tid * 10];
            const float* pe = &spe[tid * 10];
            float p[10], q[10];
#pragma unroll
            for (int k = 0; k < 10; ++k) { p[k] = pt[k]; q[k] = pe[k]; }

            float score = 0.0f;
#pragma unroll
            for (int k = 0; k < 10; ++k) score = fmaf(p[k], (float)(k + 1), score);

            float var = 0.0f;
#pragma unroll
            for (int k = 0; k < 10; ++k) {
                const float d = (float)(k + 1) - score;
                var = fmaf(p[k], d * d, var);
            }

            float cpt = 0.0f, cpe = 0.0f, acc = 0.0f;
#pragma unroll
            for (int k = 0; k < 10; ++k) {
                cpt += p[k];
                cpe += q[k];
                const float d = cpe - cpt;
                acc = fmaf(d, d, acc);
            }
            const float emd = sqrtf(acc * 0.1f);
            lacc += emd / var;   // emd * weight, weight = 1/var
        }
        __syncthreads();   // protect LDS before next tile's staging
    }

    // wave32 reduction (warpSize == 32 on gfx1250)
#pragma unroll
    for (int off = 16; off > 0; off >>= 1) lacc += __shfl_xor(lacc, off, 32);
    const int wave = tid >> 5;
    const int lane = tid & 31;
    if (lane == 0) swave[wave] = lacc;
    __syncthreads();
    if (tid == 0) {
        float s = 0.0f;
#pragma unroll
        for (int i = 0; i < 8; ++i) s += swave[i];
        partials[blockIdx.x] = s;
    }
}

__global__ __launch_bounds__(256) void emd_final_kernel(
    const float* __restrict__ partials, int n, float* __restrict__ out,
    double invB) {
    __shared__ double sd[256];
    double acc = 0.0;
    for (int i = threadIdx.x; i < n; i += 256) acc += (double)partials[i];
    sd[threadIdx.x] = acc;
    __syncthreads();
#pragma unroll
    for (int s = 128; s > 0; s >>= 1) {
        if ((int)threadIdx.x < s) sd[threadIdx.x] += sd[threadIdx.x + s];
        __syncthreads();
    }
    if (threadIdx.x == 0) out[0] = (float)(sd[0] * invB);
}

extern "C" void kernel_launch(void* const* d_in, const int* in_sizes, int n_in,
                              void* d_out, int out_size, void* d_ws, size_t ws_size,
                              hipStream_t stream) {
    const float* pt = (const float*)d_in[0];
    const float* pe = (const float*)d_in[1];
    const int totalF = in_sizes[0];
    const int B = totalF / 10;
    const int numTiles = (B + TILE_ROWS - 1) / TILE_ROWS;
    const int nblocks = numTiles < MAX_BLOCKS ? numTiles : MAX_BLOCKS;

    float* partials = (float*)d_ws;
    emd_partial_kernel<<<nblocks, TILE_ROWS, 0, stream>>>(pt, pe, partials, B, numTiles);
    emd_final_kernel<<<1, 256, 0, stream>>>(partials, nblocks, (float*)d_out,
                                            1.0 / (double)B);
}